// EncNP_41188736369123
// MI455X (gfx1250) — compile-verified
//
#include <hip/hip_runtime.h>
#include <math.h>

typedef __attribute__((ext_vector_type(2))) float v2f;
typedef __attribute__((ext_vector_type(8))) float v8f;
typedef __attribute__((ext_vector_type(4))) unsigned int v4u;
typedef __attribute__((ext_vector_type(8))) int v8i;
typedef __attribute__((ext_vector_type(4))) int v4i;

#define K_NB 24
#define ALPHA_C 100.0f
#define BETA_C 1000.0f
#define TWO_PI_C 6.28318530717958647692f

// ---------------------------------------------------------------------------
// LDS byte offset of a __shared__ object: the LDS aperture maps the low 32
// flat-address bits directly onto the LDS byte offset (ISA 10.2 aperture map).
// ---------------------------------------------------------------------------
__device__ __forceinline__ unsigned int lds_off_u32(const void* p) {
  return (unsigned int)(uintptr_t)p;
}

// ---------------------------------------------------------------------------
// Tensor Data Mover: contiguous 1-D copy of nelem f32 from global to LDS.
// D# per CDNA5 ISA:
//   group0: count=1 | lds_addr | global_addr[56:0] | type=2
//   group1: data_size=4B; tensor_dim0=tile_dim0=nelem; tensor_dim1=tile_dim1=1;
//           tensor_dim0_stride=nelem
// Issue from ONE wave only (TENSORcnt is per-wave), then s_wait_tensorcnt 0.
// ---------------------------------------------------------------------------
__device__ __forceinline__ void tdm_load_1d_f32(const float* gsrc, unsigned int lds_byte,
                                                unsigned int nelem) {
  unsigned long long ga = (unsigned long long)(uintptr_t)gsrc;
  v4u g0;
  g0[0] = 1u;                                          // count=1, user descriptor
  g0[1] = lds_byte;                                    // lds_addr (bytes)
  g0[2] = (unsigned int)ga;                            // global_addr[31:0]
  g0[3] = (unsigned int)((ga >> 32) & 0x1FFFFFFu) | (2u << 30);  // addr[56:32]|type=2
  v8i g1;
  g1[0] = 0x00020000;                                  // data_size=2 (4 bytes)
  g1[1] = (int)((nelem & 0xFFFFu) << 16);              // tensor_dim0[15:0]
  g1[2] = (int)((nelem >> 16) | (1u << 16));           // tensor_dim0[31:16], tensor_dim1=1
  g1[3] = (int)((nelem & 0xFFFFu) << 16);              // tile_dim0 (= nelem, <= 65535)
  g1[4] = 1;                                           // tile_dim1=1, tile_dim2=0
  g1[5] = (int)nelem;                                  // tensor_dim0_stride[31:0]
  g1[6] = 0;                                           // stride0[47:32], stride1[15:0]
  g1[7] = 0;                                           // stride1[47:16]
  v4i g2 = {};                                         // 2-D tensor: groups 2/3 unused
  v4i g3 = {};
#if __has_include(<hip/amd_detail/amd_gfx1250_TDM.h>)
  v8i g4 = {};
  __builtin_amdgcn_tensor_load_to_lds(g0, g1, g2, g3, g4, 0);
#else
  __builtin_amdgcn_tensor_load_to_lds(g0, g1, g2, g3, 0);
#endif
}

// ---------------------------------------------------------------------------
// Block-wide sum reduction (256 threads, wave32). Result valid on tid 0.
// ---------------------------------------------------------------------------
__device__ __forceinline__ float block_reduce_sum(float v, float* s) {
  int tid = threadIdx.x, lane = tid & 31, w = tid >> 5;
#pragma unroll
  for (int off = 16; off > 0; off >>= 1) v += __shfl_down(v, off, 32);
  if (lane == 0) s[w] = v;
  __syncthreads();
  float r = 0.f;
  if (tid < (int)(blockDim.x >> 5)) r = s[tid];
  if (w == 0) {
#pragma unroll
    for (int off = 16; off > 0; off >>= 1) r += __shfl_down(r, off, 32);
  }
  __syncthreads();
  return r;
}

// ---------------------------------------------------------------------------
// Furthest point sampling. One block per batch. TDM bulk-copies the batch's
// point cloud (AoS, stride-3 -> bank-conflict-free since gcd(3,64)=1) into
// LDS; per-thread running min-distances live in registers. Serial loop is the
// latency bottleneck of the whole net, so everything it touches is on-WGP.
// ---------------------------------------------------------------------------
__global__ __launch_bounds__(1024) void fps_kernel(const float* __restrict__ xyz,
                                                   int* __restrict__ fi, int N, int M) {
  __shared__ float sxyz[4096 * 3];
  __shared__ float wmax[32];
  __shared__ int widx[32];
  __shared__ int sfar;
  const int b = blockIdx.x, tid = threadIdx.x;
  const int nthr = 1024;
  const int npt = N / nthr;  // 4 (N=4096) or 2 (N=2048)
  float dloc[4];
  if (tid < 32) {  // wave 0 issues the DMA and waits on its TENSORcnt
    tdm_load_1d_f32(xyz + (size_t)b * N * 3, lds_off_u32(sxyz), (unsigned)(N * 3));
    __builtin_amdgcn_s_wait_tensorcnt(0);
  }
  for (int j = 0; j < npt; ++j) dloc[j] = 1e10f;
  if (tid == 0) sfar = 0;
  __syncthreads();
  for (int it = 0; it < M; ++it) {
    int far = sfar;
    if (tid == 0) fi[(size_t)b * M + it] = far;
    float cx = sxyz[3 * far + 0], cy = sxyz[3 * far + 1], cz = sxyz[3 * far + 2];
    float bmax = -1.0f;
    int bidx = 0;
    for (int j = 0; j < npt; ++j) {
      int i = tid + j * nthr;
      float dx = sxyz[3 * i + 0] - cx;
      float dy = sxyz[3 * i + 1] - cy;
      float dz = sxyz[3 * i + 2] - cz;
      float d = dx * dx + dy * dy + dz * dz;
      float nd = fminf(dloc[j], d);
      dloc[j] = nd;
      if (nd > bmax) { bmax = nd; bidx = i; }
    }
#pragma unroll
    for (int off = 16; off > 0; off >>= 1) {
      float om = __shfl_down(bmax, off, 32);
      int oi = __shfl_down(bidx, off, 32);
      if (om > bmax) { bmax = om; bidx = oi; }
    }
    if ((tid & 31) == 0) { wmax[tid >> 5] = bmax; widx[tid >> 5] = bidx; }
    __syncthreads();
    if (tid == 0) {
      float m = -1.0f;
      int mi = 0;
      for (int w = 0; w < 32; ++w)
        if (wmax[w] > m) { m = wmax[w]; mi = widx[w]; }
      sfar = mi;
    }
    __syncthreads();
  }
}

// ---------------------------------------------------------------------------
// Gather sampled centers: lc_xyz [B,G,3], lc_x [B,G,C] from x [B,C,N].
// ---------------------------------------------------------------------------
__global__ void gather_kernel(const float* __restrict__ xyz, const float* __restrict__ x,
                              const int* __restrict__ fi, float* __restrict__ lc_xyz,
                              float* __restrict__ lc_x, int N, int C, int G) {
  int b = blockIdx.y, g = blockIdx.x;
  int idx = fi[(size_t)b * G + g];
  for (int c = threadIdx.x; c < C; c += blockDim.x)
    lc_x[((size_t)b * G + g) * C + c] = x[((size_t)b * C + c) * N + idx];
  if (threadIdx.x < 3)
    lc_xyz[((size_t)b * G + g) * 3 + threadIdx.x] =
        xyz[((size_t)b * N + idx) * 3 + threadIdx.x];
}

// ---------------------------------------------------------------------------
// KNN via V_WMMA_F32_16X16X4_F32, B-operand fed from an LDS-resident point
// cloud bulk-loaded once per block by the TDM (removes the per-chunk
// global_load + s_wait_loadcnt stall that fed the WMMA in the previous rev).
// A = [-2cx,-2cy,-2cz,|c|^2] (16x4, K padded 3->4), B = [px,py,pz,1]^T,
// C = |p|^2 broadcast  =>  D = full squared distance in one WMMA.
// ---------------------------------------------------------------------------
__global__ __launch_bounds__(256) void knn_kernel(const float* __restrict__ xyz,
                                                  const float* __restrict__ centers,
                                                  int* __restrict__ knn_idx, int N, int G) {
  __shared__ float sxyz[4096 * 3];  // 48 KB of the WGP's 320 KB LDS
  __shared__ float distT[16][132];  // stride 132: conflict-free column reads
  __shared__ float topd[16][K_NB];
  __shared__ int topi[16][K_NB];
  int b = blockIdx.y;
  int g0 = blockIdx.x * 16;
  int tid = threadIdx.x;
  int wave = tid >> 5;
  int lane = tid & 31;
  int half = lane & 15;

  if (tid < 32) {  // wave 0: DMA the whole batch cloud into LDS
    tdm_load_1d_f32(xyz + (size_t)b * N * 3, lds_off_u32(sxyz), (unsigned)(N * 3));
    __builtin_amdgcn_s_wait_tensorcnt(0);
  }

  const float* c = centers + ((size_t)b * G + g0 + half) * 3;
  float cx = c[0], cy = c[1], cz = c[2];
  v2f A;
  if (lane < 16) { A.x = -2.f * cx; A.y = -2.f * cy; }               // K=0, K=1
  else           { A.x = -2.f * cz; A.y = cx * cx + cy * cy + cz * cz; }  // K=2, K=3

  if (tid < 16) {
    for (int j = 0; j < K_NB; ++j) { topd[tid][j] = 3.4e38f; topi[tid][j] = 0; }
  }
  __syncthreads();

  for (int nbase = 0; nbase < N; nbase += 128) {
    int n = nbase + wave * 16 + half;
    float px = sxyz[3 * n + 0], py = sxyz[3 * n + 1], pz = sxyz[3 * n + 2];
    v2f Bm;
    if (lane < 16) { Bm.x = px; Bm.y = py; }   // K=0, K=1
    else           { Bm.x = pz; Bm.y = 1.0f; } // K=2, K=3
    float pn2 = px * px + py * py + pz * pz;
    v8f Cc;
#pragma unroll
    for (int r = 0; r < 8; ++r) Cc[r] = pn2;   // ||p||^2 per column, all rows
    v8f D = __builtin_amdgcn_wmma_f32_16x16x4_f32(false, A, false, Bm, (short)0, Cc,
                                                  false, false);
    int col = wave * 16 + half;
    int rbase = (lane < 16) ? 0 : 8;
#pragma unroll
    for (int r = 0; r < 8; ++r) distT[rbase + r][col] = D[r];
    __syncthreads();
    if (tid < 16) {
      for (int j = 0; j < 128; ++j) {
        float d = distT[tid][j];
        if (d < topd[tid][K_NB - 1]) {
          int idx = nbase + j;
          int pos = K_NB - 1;
          while (pos > 0 && topd[tid][pos - 1] > d) {
            topd[tid][pos] = topd[tid][pos - 1];
            topi[tid][pos] = topi[tid][pos - 1];
            --pos;
          }
          topd[tid][pos] = d;
          topi[tid][pos] = idx;
        }
      }
    }
    __syncthreads();
  }
  if (tid < 16)
    for (int j = 0; j < K_NB; ++j)
      knn_idx[((size_t)b * G + g0 + tid) * K_NB + j] = topi[tid][j];
}

// ---------------------------------------------------------------------------
// Global sum / sum-sq for the two normalize() stds (ddof=1 over ALL elements).
// ---------------------------------------------------------------------------
__global__ __launch_bounds__(256) void norm_stats_kernel(
    const float* __restrict__ xyz, const float* __restrict__ x,
    const float* __restrict__ lc_xyz, const float* __restrict__ lc_x,
    const int* __restrict__ knn, float* __restrict__ stats, int N, int C, int G) {
  int b = blockIdx.y, g = blockIdx.x, tid = threadIdx.x;
  __shared__ int sidx[K_NB];
  __shared__ float lc[3];
  __shared__ float s[32];
  if (tid < K_NB) sidx[tid] = knn[((size_t)b * G + g) * K_NB + tid];
  if (tid < 3) lc[tid] = lc_xyz[((size_t)b * G + g) * 3 + tid];
  __syncthreads();
  float sx = 0.f, ssx = 0.f, sz = 0.f, ssz = 0.f;
  for (int t = tid; t < K_NB * C; t += 256) {
    int k = t / C, cc = t - (t / C) * C;
    float d = x[((size_t)b * C + cc) * N + sidx[k]] - lc_x[((size_t)b * G + g) * C + cc];
    sx += d;
    ssx += d * d;
  }
  for (int t = tid; t < K_NB * 3; t += 256) {
    int k = t / 3, j = t - (t / 3) * 3;
    float d = xyz[((size_t)b * N + sidx[k]) * 3 + j] - lc[j];
    sz += d;
    ssz += d * d;
  }
  sx = block_reduce_sum(sx, s);
  ssx = block_reduce_sum(ssx, s);
  sz = block_reduce_sum(sz, s);
  ssz = block_reduce_sum(ssz, s);
  if (tid == 0) {
    atomicAdd(&stats[0], sx);
    atomicAdd(&stats[1], ssx);
    atomicAdd(&stats[2], sz);
    atomicAdd(&stats[3], ssz);
  }
}

// ---------------------------------------------------------------------------
// Fused LGA + k_anp pooling. k_anp's weight e is constant along K, so
// pooled = 2 * mean_K((knn_x_w + pe) * pe); w is never materialized.
// ---------------------------------------------------------------------------
__global__ __launch_bounds__(256) void lga_pool_kernel(
    const float* __restrict__ xyz, const float* __restrict__ x,
    const float* __restrict__ lc_xyz, const float* __restrict__ lc_x,
    const int* __restrict__ knn, const float* __restrict__ Bmat,
    const float* __restrict__ stats, float* __restrict__ pooled, int N, int C, int G) {
  int b = blockIdx.y, g = blockIdx.x, tid = threadIdx.x;
  const int C2 = 2 * C, half = C / 2, F = C / 3, twoF = 2 * F;
  __shared__ float embin[K_NB][8];
  __shared__ float sknx[K_NB][144];
  __shared__ float lcx[144];
  __shared__ float lc[3];
  __shared__ int sidx[K_NB];
  __shared__ float Bs[7 * 72];
  __shared__ float invde[48];
  __shared__ float istd[2];

  if (tid < K_NB) sidx[tid] = knn[((size_t)b * G + g) * K_NB + tid];
  if (tid < 3) lc[tid] = lc_xyz[((size_t)b * G + g) * 3 + tid];
  for (int t = tid; t < C; t += 256) lcx[t] = lc_x[((size_t)b * G + g) * C + t];
  for (int t = tid; t < 7 * half; t += 256) Bs[t] = Bmat[t];
  for (int t = tid; t < F; t += 256)
    invde[t] = BETA_C * __powf(ALPHA_C, -(float)t / (float)F);
  if (tid == 0) {
    float nx = 4.f * (float)G * K_NB * (float)C;
    float nz = 4.f * (float)G * K_NB * 3.f;
    float mx = stats[0] / nx;
    float vx = (stats[1] - nx * mx * mx) / (nx - 1.f);
    float mz = stats[2] / nz;
    float vz = (stats[3] - nz * mz * mz) / (nz - 1.f);
    istd[0] = 1.f / (sqrtf(fmaxf(vx, 0.f)) + 1e-5f);
    istd[1] = 1.f / (sqrtf(fmaxf(vz, 0.f)) + 1e-5f);
  }
  __syncthreads();
  if (tid < K_NB) {
    int idx = sidx[tid];
    float ax = (xyz[((size_t)b * N + idx) * 3 + 0] - lc[0]) * istd[1];
    float ay = (xyz[((size_t)b * N + idx) * 3 + 1] - lc[1]) * istd[1];
    float az = (xyz[((size_t)b * N + idx) * 3 + 2] - lc[2]) * istd[1];
    embin[tid][0] = ax;
    embin[tid][1] = ay;
    embin[tid][2] = az;
    embin[tid][3] = ay * lc[2] - az * lc[1];  // cross(knn_n, lc)
    embin[tid][4] = az * lc[0] - ax * lc[2];
    embin[tid][5] = ax * lc[1] - ay * lc[0];
    embin[tid][6] = ax * lc[0] + ay * lc[1] + az * lc[2];  // dot
    embin[tid][7] = 0.f;
  }
  for (int t = tid; t < K_NB * C; t += 256) {
    int k = t / C, cc = t - (t / C) * C;
    sknx[k][cc] = (x[((size_t)b * C + cc) * N + sidx[k]] - lcx[cc]) * istd[0];
  }
  __syncthreads();

  for (int ch = tid; ch < C2; ch += 256) {
    int c3 = ch / twoF;
    int j = ch - c3 * twoF;
    int f = j >> 1;
    int isCos = j & 1;
    float inv = invde[f];
    float dl = lc[c3] * inv;
    float pel = isCos ? __cosf(dl) : __sinf(dl);
    float acc = 0.f;
    for (int k = 0; k < K_NB; ++k) {
      float dk = embin[k][c3] * inv;
      float pe = (isCos ? __cosf(dk) : __sinf(dk)) + pel;
      float vx;
      if (ch < C) {
        vx = sknx[k][ch];
      } else {
        int ce = ch - C;
        int col = (ce < half) ? ce : ce - half;
        float pr = 0.f;
#pragma unroll
        for (int q = 0; q < 7; ++q) pr += embin[k][q] * Bs[q * half + col];
        pr *= TWO_PI_C;
        float fv = (ce < half) ? __sinf(pr) : __cosf(pr);
        float f2 = fv * fv;
        vx = fv * f2 * f2;  // sign(f)*|f|^5 == f^5
      }
      acc += (vx + pe) * pe;
    }
    pooled[((size_t)b * C2 + ch) * G + g] = 2.f * acc * (1.0f / K_NB);
  }
}

// ---------------------------------------------------------------------------
// BatchNorm batch statistics (training mode, biased var) over (B, G).
// ---------------------------------------------------------------------------
__global__ __launch_bounds__(256) void bn_stats_kernel(const float* __restrict__ pooled,
                                                       float* __restrict__ mean,
                                                       float* __restrict__ var, int C2,
                                                       int G) {
  int ch = blockIdx.x, tid = threadIdx.x;
  __shared__ float s[32];
  float su = 0.f, ss = 0.f;
  int BG = 4 * G;
  for (int e = tid; e < BG; e += 256) {
    int b = e / G, g = e - b * G;
    float v = pooled[((size_t)b * C2 + ch) * G + g];
    su += v;
    ss += v * v;
  }
  su = block_reduce_sum(su, s);
  ss = block_reduce_sum(ss, s);
  if (tid == 0) {
    float m = su / (float)BG;
    mean[ch] = m;
    var[ch] = ss / (float)BG - m * m;
  }
}

// ---------------------------------------------------------------------------
// BN normalize + exact GELU.
// ---------------------------------------------------------------------------
__global__ void bn_gelu_kernel(const float* __restrict__ pooled,
                               const float* __restrict__ mean,
                               const float* __restrict__ var,
                               const float* __restrict__ gamma,
                               const float* __restrict__ beta, float* __restrict__ out,
                               int C2, int G, int total) {
  int i = blockIdx.x * blockDim.x + threadIdx.x;
  if (i >= total) return;
  int ch = (i / G) % C2;
  float y = (pooled[i] - mean[ch]) * rsqrtf(var[ch] + 1e-5f) * gamma[ch] + beta[ch];
  out[i] = 0.5f * y * (1.f + erff(y * 0.70710678118654752f));
}

// ---------------------------------------------------------------------------
extern "C" void kernel_launch(void* const* d_in, const int* in_sizes, int n_in,
                              void* d_out, int out_size, void* d_ws, size_t ws_size,
                              hipStream_t stream) {
  (void)in_sizes; (void)n_in; (void)out_size; (void)ws_size;
  const float* xyz = (const float*)d_in[0];
  const float* x0  = (const float*)d_in[1];
  const float* B0  = (const float*)d_in[2];
  const float* B1  = (const float*)d_in[3];
  const float* ga0 = (const float*)d_in[4];
  const float* be0 = (const float*)d_in[5];
  const float* ga1 = (const float*)d_in[6];
  const float* be1 = (const float*)d_in[7];
  float* out = (float*)d_out;

  const int B = 4, N1 = 4096, C0 = 72;
  const int G1 = N1 / 2, G2 = N1 / 4;

  char* ws = (char*)d_ws;
  size_t off = 0;
  auto take = [&](size_t bytes) {
    char* p = ws + off;
    off += (bytes + 255) & ~(size_t)255;
    return p;
  };
  float* stats   = (float*)take(8 * sizeof(float));
  int*   fi      = (int*)take((size_t)B * G1 * sizeof(int));
  float* lcxyz   = (float*)take((size_t)B * G1 * 3 * sizeof(float));
  float* lcx     = (float*)take((size_t)B * G1 * 144 * sizeof(float));
  int*   knn     = (int*)take((size_t)B * G1 * K_NB * sizeof(int));
  float* pooled  = (float*)take((size_t)B * 288 * G1 * sizeof(float));
  float* x2      = (float*)take((size_t)B * 144 * G1 * sizeof(float));
  float* xyz2    = (float*)take((size_t)B * G1 * 3 * sizeof(float));
  float* bn_mean = (float*)take(288 * sizeof(float));
  float* bn_var  = (float*)take(288 * sizeof(float));

  // -------- stage 1: N=4096, C=72, G=2048 --------
  {
    int N = N1, C = C0, G = G1, C2 = 2 * C;
    fps_kernel<<<B, 1024, 0, stream>>>(xyz, fi, N, G);
    gather_kernel<<<dim3(G, B), 128, 0, stream>>>(xyz, x0, fi, xyz2, lcx, N, C, G);
    knn_kernel<<<dim3(G / 16, B), 256, 0, stream>>>(xyz, xyz2, knn, N, G);
    hipMemsetAsync(stats, 0, 4 * sizeof(float), stream);
    norm_stats_kernel<<<dim3(G, B), 256, 0, stream>>>(xyz, x0, xyz2, lcx, knn, stats, N,
                                                      C, G);
    lga_pool_kernel<<<dim3(G, B), 256, 0, stream>>>(xyz, x0, xyz2, lcx, knn, B0, stats,
                                                    pooled, N, C, G);
    bn_stats_kernel<<<C2, 256, 0, stream>>>(pooled, bn_mean, bn_var, C2, G);
    int total = B * C2 * G;
    bn_gelu_kernel<<<(total + 255) / 256, 256, 0, stream>>>(pooled, bn_mean, bn_var, ga0,
                                                            be0, x2, C2, G, total);
  }
  // -------- stage 2: N=2048, C=144, G=1024 --------
  {
    int N = G1, C = 2 * C0, G = G2, C2 = 2 * C;
    fps_kernel<<<B, 1024, 0, stream>>>(xyz2, fi, N, G);
    gather_kernel<<<dim3(G, B), 256, 0, stream>>>(xyz2, x2, fi, lcxyz, lcx, N, C, G);
    knn_kernel<<<dim3(G / 16, B), 256, 0, stream>>>(xyz2, lcxyz, knn, N, G);
    hipMemsetAsync(stats, 0, 4 * sizeof(float), stream);
    norm_stats_kernel<<<dim3(G, B), 256, 0, stream>>>(xyz2, x2, lcxyz, lcx, knn, stats,
                                                      N, C, G);
    lga_pool_kernel<<<dim3(G, B), 256, 0, stream>>>(xyz2, x2, lcxyz, lcx, knn, B1, stats,
                                                    pooled, N, C, G);
    bn_stats_kernel<<<C2, 256, 0, stream>>>(pooled, bn_mean, bn_var, C2, G);
    int total = B * C2 * G;
    bn_gelu_kernel<<<(total + 255) / 256, 256, 0, stream>>>(pooled, bn_mean, bn_var, ga1,
                                                            be1, out, C2, G, total);
  }
}